// GraphTranslator_53850299958028
// MI455X (gfx1250) — compile-verified
//
#include <hip/hip_runtime.h>
#include <math.h>

#define DIM 128
#define EPSF 1e-16f
#define APAD 132   // LDS row stride (floats): 132 mod 64 = 4 -> conflict-free b64 A loads

typedef float v2f __attribute__((ext_vector_type(2)));
typedef float v8f __attribute__((ext_vector_type(8)));

// ---- monotonic float <-> uint encoding so segment-max = global_atomic_max_u32
__device__ __forceinline__ unsigned f2ord(float f) {
    unsigned u = __float_as_uint(f);
    return (u & 0x80000000u) ? ~u : (u | 0x80000000u);
}
__device__ __forceinline__ float ord2f(unsigned u) {
    return (u & 0x80000000u) ? __uint_as_float(u & 0x7FFFFFFFu)
                             : __uint_as_float(~u);
}
#define ORD_NEG_INF 0x007FFFFFu   // f2ord(-inf)

// ======================================================================
// Fused 4-way GEMM: Yi[nrows,128] = X[nrows,128] @ Wi[128,128] + bi  (i=0..3)
// Block = 256 threads = 8 waves. Block stages one 16x128 A tile in LDS
// (coalesced, bank-pad APAD); each wave owns one 16-col tile and runs the
// K-loop once, issuing 4 v_wmma_f32_16x16x4_f32 per A fragment.
// ======================================================================
__global__ __launch_bounds__(256) void gemm128x4_wmma(
    const float* __restrict__ X,
    const float* __restrict__ W0, const float* __restrict__ b0, float* __restrict__ Y0,
    const float* __restrict__ W1, const float* __restrict__ b1, float* __restrict__ Y1,
    const float* __restrict__ W2, const float* __restrict__ b2, float* __restrict__ Y2,
    const float* __restrict__ W3, const float* __restrict__ b3, float* __restrict__ Y3,
    int nrows)
{
    __shared__ float As[16 * APAD];

    const int tid  = threadIdx.x;
    const int lane = tid & 31;
    const int wid  = tid >> 5;
    const int m0   = blockIdx.x * 16;

    // ---- cooperative A-tile stage: 16x128 floats = 512 float4, 2 per thread
    #pragma unroll
    for (int i = tid; i < 512; i += 256) {
        const int row = i >> 3;            // 8 float4 per 128-float row
        const int c4  = (i & 7) * 4;
        int m = m0 + row; if (m >= nrows) m = nrows - 1;   // clamp keeps EXEC full
        const float4 v = *(const float4*)(X + (size_t)m * DIM + c4);
        float* dp = As + row * APAD + c4;
        dp[0] = v.x; dp[1] = v.y; dp[2] = v.z; dp[3] = v.w;
    }
    __syncthreads();

    // ---- WMMA operand addressing (ISA 7.12.2 fp32 layouts)
    const int half = lane >> 4;            // 0: K{0,1}/rows 0-7 ; 1: K{2,3}/rows 8-15
    const int l15  = lane & 15;
    const int n0   = wid * 16;             // this wave's 16-col tile

    const float* ap  = As + l15 * APAD + 2 * half;               // A fragment (LDS)
    const size_t wo  = (size_t)(2 * half) * DIM + n0 + l15;      // B fragment base
    const float* wp0 = W0 + wo;
    const float* wp1 = W1 + wo;
    const float* wp2 = W2 + wo;
    const float* wp3 = W3 + wo;

    v8f a0 = {}, a1 = {}, a2 = {}, a3 = {};
    #pragma unroll 4
    for (int k = 0; k < DIM; k += 4) {
        v2f a;  a.x  = ap[0];   a.y  = ap[1];        // ds_load_b64, conflict-free
        v2f q0; q0.x = wp0[0];  q0.y = wp0[DIM];
        v2f q1; q1.x = wp1[0];  q1.y = wp1[DIM];
        v2f q2; q2.x = wp2[0];  q2.y = wp2[DIM];
        v2f q3; q3.x = wp3[0];  q3.y = wp3[DIM];
        a0 = __builtin_amdgcn_wmma_f32_16x16x4_f32(false, a, false, q0, (short)0, a0, false, false);
        a1 = __builtin_amdgcn_wmma_f32_16x16x4_f32(false, a, false, q1, (short)0, a1, false, false);
        a2 = __builtin_amdgcn_wmma_f32_16x16x4_f32(false, a, false, q2, (short)0, a2, false, false);
        a3 = __builtin_amdgcn_wmma_f32_16x16x4_f32(false, a, false, q3, (short)0, a3, false, false);
        ap  += 4;
        wp0 += 4 * DIM; wp1 += 4 * DIM; wp2 += 4 * DIM; wp3 += 4 * DIM;
    }

    // ---- epilogue: bias + store (C/D layout: VGPR r -> M = r + 8*half, N = l15)
    const int   nc    = n0 + l15;
    const int   mbase = m0 + 8 * half;
    const float c0 = b0[nc], c1 = b1[nc], c2 = b2[nc], c3 = b3[nc];
    #pragma unroll
    for (int r = 0; r < 8; ++r) {
        const int m = mbase + r;
        if (m < nrows) {
            const size_t o = (size_t)m * DIM + nc;
            Y0[o] = a0[r] + c0;
            Y1[o] = a1[r] + c1;
            Y2[o] = a2[r] + c2;
            Y3[o] = a3[r] + c3;
        }
    }
}

// ======================================================================
// Edge attention logits: one wave per edge, lane covers 4 channels.
// alpha[e,h] = dot(Q[dst,h,:], K[src,h,:]) * scale ; atomic segment-max.
// ======================================================================
template<int H>
__global__ __launch_bounds__(256) void edge_logits(
    const float* __restrict__ Q, const float* __restrict__ K,
    const long long* __restrict__ ei, int E,
    float* __restrict__ alpha, unsigned* __restrict__ Menc, float scale)
{
    const int C = DIM / H;        // channels per head
    const int G = C / 4;          // lanes per head group
    const int lane = threadIdx.x & 31;
    const int wid  = threadIdx.x >> 5;
    const long long e = (long long)blockIdx.x * 8 + wid;
    if (e >= E) return;

    const int src = (int)ei[e];
    const int dst = (int)ei[(long long)E + e];

    const float4 qv = ((const float4*)(Q + (size_t)dst * DIM))[lane];
    const float4 kv = ((const float4*)(K + (size_t)src * DIM))[lane];
    float acc = qv.x * kv.x + qv.y * kv.y + qv.z * kv.z + qv.w * kv.w;

    #pragma unroll
    for (int off = 1; off < G; off <<= 1)
        acc += __shfl_xor(acc, off, 32);

    if ((lane & (G - 1)) == 0) {
        const int head = lane / G;
        const float a = acc * scale;
        alpha[(size_t)e * H + head] = a;
        atomicMax(&Menc[(size_t)dst * H + head], f2ord(a));
    }
}

// ======================================================================
// ea = exp(alpha - m[dst]) ; denom[dst] += ea   (one thread per edge*head)
// ======================================================================
template<int H>
__global__ void edge_exp(const long long* __restrict__ ei, int E,
                         float* __restrict__ alpha,
                         const unsigned* __restrict__ Menc,
                         float* __restrict__ Den)
{
    const long long tid = (long long)blockIdx.x * blockDim.x + threadIdx.x;
    if (tid >= (long long)E * H) return;
    const long long e = tid / H;                 // constant H -> strength-reduced
    const int h = (int)(tid - e * H);
    const int dst = (int)ei[(long long)E + e];

    float m = ord2f(Menc[(size_t)dst * H + h]);
    if (!__builtin_isfinite(m)) m = 0.0f;        // match reference's isfinite guard
    const float ea = __expf(alpha[tid] - m);
    alpha[tid] = ea;
    atomicAdd(&Den[(size_t)dst * H + h], ea);
}

// ======================================================================
// Out[dst] += (ea / (denom[dst]+eps)) * V[src]   one wave per edge
// ======================================================================
template<int H>
__global__ __launch_bounds__(256) void edge_agg(
    const float* __restrict__ V, const long long* __restrict__ ei, int E,
    const float* __restrict__ alpha, const float* __restrict__ Den,
    float* __restrict__ Out)
{
    const int C = DIM / H;
    const int G = C / 4;
    const int lane = threadIdx.x & 31;
    const int wid  = threadIdx.x >> 5;
    const long long e = (long long)blockIdx.x * 8 + wid;
    if (e >= E) return;

    const int src = (int)ei[e];
    const int dst = (int)ei[(long long)E + e];
    const int head = lane / G;

    const float w = alpha[(size_t)e * H + head] /
                    (Den[(size_t)dst * H + head] + EPSF);
    const float4 vv = ((const float4*)(V + (size_t)src * DIM))[lane];
    float* op = Out + (size_t)dst * DIM + lane * 4;
    atomicAdd(op + 0, w * vv.x);
    atomicAdd(op + 1, w * vv.y);
    atomicAdd(op + 2, w * vv.z);
    atomicAdd(op + 3, w * vv.w);
}

__global__ void init_seg(unsigned* __restrict__ Menc, float* __restrict__ Den,
                         int count)
{
    const int i = blockIdx.x * blockDim.x + threadIdx.x;
    if (i < count) { Menc[i] = ORD_NEG_INF; Den[i] = 0.0f; }
}

// ======================================================================
extern "C" void kernel_launch(void* const* d_in, const int* in_sizes, int n_in,
                              void* d_out, int out_size, void* d_ws, size_t ws_size,
                              hipStream_t stream)
{
    const float*     x  = (const float*)d_in[0];
    const long long* ei = (const long long*)d_in[1];
    const float *Wq1 = (const float*)d_in[2],  *bq1 = (const float*)d_in[3];
    const float *Wk1 = (const float*)d_in[4],  *bk1 = (const float*)d_in[5];
    const float *Wv1 = (const float*)d_in[6],  *bv1 = (const float*)d_in[7];
    const float *Ws1 = (const float*)d_in[8],  *bs1 = (const float*)d_in[9];
    const float *Wq2 = (const float*)d_in[10], *bq2 = (const float*)d_in[11];
    const float *Wk2 = (const float*)d_in[12], *bk2 = (const float*)d_in[13];
    const float *Wv2 = (const float*)d_in[14], *bv2 = (const float*)d_in[15];
    const float *Ws2 = (const float*)d_in[16], *bs2 = (const float*)d_in[17];

    const int N = in_sizes[0] / DIM;
    const int E = in_sizes[1] / 2;
    float* out = (float*)d_out;

    // workspace layout (Q/K/V + alpha + seg-stats reused across both layers)
    const size_t ND = (size_t)N * DIM;
    float*    Q     = (float*)d_ws;
    float*    Kb    = Q  + ND;
    float*    Vb    = Kb + ND;
    float*    Hh    = Vb + ND;                 // layer-1 output (skip + attn)
    float*    alpha = Hh + ND;                 // E*8 floats (layer 2 uses E of them)
    unsigned* Menc  = (unsigned*)(alpha + (size_t)E * 8);
    float*    Den   = (float*)(Menc + (size_t)N * 8);

    const dim3 blk(256);
    const int gemmBlocks = (N + 15) / 16;      // one 16-row tile per block
    const int edgeBlocks = (E + 7) / 8;        // 8 waves (edges) per block

    // -------- layer 1 (heads=8, C=16) --------
    gemm128x4_wmma<<<gemmBlocks, blk, 0, stream>>>(x,
        Wq1, bq1, Q,  Wk1, bk1, Kb,  Wv1, bv1, Vb,  Ws1, bs1, Hh, N);
    init_seg<<<(N * 8 + 255) / 256, blk, 0, stream>>>(Menc, Den, N * 8);
    edge_logits<8><<<edgeBlocks, blk, 0, stream>>>(Q, Kb, ei, E, alpha, Menc, 0.25f);
    edge_exp<8><<<(int)(((long long)E * 8 + 255) / 256), blk, 0, stream>>>(ei, E, alpha, Menc, Den);
    edge_agg<8><<<edgeBlocks, blk, 0, stream>>>(Vb, ei, E, alpha, Den, Hh);

    // -------- layer 2 (heads=1, C=128) --------
    gemm128x4_wmma<<<gemmBlocks, blk, 0, stream>>>(Hh,
        Wq2, bq2, Q,  Wk2, bk2, Kb,  Wv2, bv2, Vb,  Ws2, bs2, out, N);
    init_seg<<<(N + 255) / 256, blk, 0, stream>>>(Menc, Den, N);
    const float scale2 = 0.08838834764831845f;  // 1/sqrt(128)
    edge_logits<1><<<edgeBlocks, blk, 0, stream>>>(Q, Kb, ei, E, alpha, Menc, scale2);
    edge_exp<1><<<(E + 255) / 256, blk, 0, stream>>>(ei, E, alpha, Menc, Den);
    edge_agg<1><<<edgeBlocks, blk, 0, stream>>>(Vb, ei, E, alpha, Den, out);
}